// MatCF_33122787786945
// MI455X (gfx1250) — compile-verified
//
#include <hip/hip_runtime.h>

typedef __attribute__((ext_vector_type(2))) float v2f;
typedef __attribute__((ext_vector_type(8))) float v8f;

namespace {
constexpr int K_DIM = 64;   // embedding dim (reference K)
}

// One wave32 produces 16 outputs via V_WMMA_F32_16X16X4_F32 accumulation.
// A = 16 gathered user rows (16x4 fp32 chunk), B = 16 gathered item columns
// (4x16 fp32 chunk); diag(D) after 16 K-chunks = the 16 dot products.
//
// Cache policy: user rows / indices are ~single-use (B/M ~ 1.05) -> NT loads,
// so the 128 MB item matrix (reused ~2.1x per column) stays resident in the
// 192 MB L2 while the ~256 MB user stream bypasses it.
__global__ __launch_bounds__(256, 4) void matcf_wmma_kernel(
    const int* __restrict__ user,
    const int* __restrict__ item,
    const float* __restrict__ user_emb,
    const float* __restrict__ item_emb,
    float* __restrict__ out,
    int n_batch,
    int n_items)
{
    const int tid  = threadIdx.x;
    const int lane = tid & 31;                       // wave32
    const int wave = tid >> 5;                       // 0..7
    const long base = ((long)blockIdx.x * 8 + wave) * 16;
    if (base >= n_batch) return;                     // uniform per wave: EXEC stays full

    const int m  = lane & 15;                        // which of the 16 outputs
    const int hi = lane >> 4;                        // K-half within a 4-chunk

    long gidx = base + m;
    if (gidx >= n_batch) gidx = n_batch - 1;         // clamp loads; store predicated below

    const int u  = __builtin_nontemporal_load(user + gidx);
    const int it = __builtin_nontemporal_load(item + gidx);

    // A stream: single base + immediate offsets (0..240 bytes)
    const float* __restrict__ urow = user_emb + (long)u * K_DIM + (hi << 1);
    // B stream: strided pointer walk, +4*N per K-chunk (no per-step 64-bit mul)
    const float* __restrict__ pcol = item_emb + (long)(hi << 1) * n_items + it;
    const long   stepN = 4L * n_items;

    v8f acc = {0.f, 0.f, 0.f, 0.f, 0.f, 0.f, 0.f, 0.f};

    #pragma unroll
    for (int k0 = 0; k0 < K_DIM; k0 += 4) {
        // A operand (16x4 f32): lane holds urow[k0+2hi], urow[k0+2hi+1] -> one NT b64 load
        v2f a = __builtin_nontemporal_load((const v2f*)(urow + k0));
        // B operand (4x16 f32): lane holds column `it` at rows k0+2hi, k0+2hi+1
        v2f b;
        b.x = pcol[0];
        b.y = pcol[n_items];
        pcol += stepN;
        // (neg_a, A, neg_b, B, c_mod, C, reuse_a, reuse_b)
        acc = __builtin_amdgcn_wmma_f32_16x16x4_f32(
            false, a, false, b, (short)0, acc, false, false);
    }

    // Extract diag(D): element (m,m) lives at lane m / vgpr m (m<8)
    // or lane m+16 / vgpr m-8 (m>=8). Owner lanes: 0-7 and 24-31.
    const int v = lane & 7;
    float d = acc[0];
    d = (v == 1) ? acc[1] : d;
    d = (v == 2) ? acc[2] : d;
    d = (v == 3) ? acc[3] : d;
    d = (v == 4) ? acc[4] : d;
    d = (v == 5) ? acc[5] : d;
    d = (v == 6) ? acc[6] : d;
    d = (v == 7) ? acc[7] : d;

    // pre = relu(4 - pre); pre = relu(4 - pre) + 1
    float p = fmaxf(4.0f - d, 0.0f);
    float q = fmaxf(4.0f - p, 0.0f) + 1.0f;

    const bool owner = (lane < 8) || (lane >= 24);
    if (owner && (base + m) < n_batch) {
        out[base + m] = q;
    }
}

extern "C" void kernel_launch(void* const* d_in, const int* in_sizes, int n_in,
                              void* d_out, int out_size, void* d_ws, size_t ws_size,
                              hipStream_t stream) {
    const int*   user     = (const int*)d_in[0];
    const int*   item     = (const int*)d_in[1];
    const float* user_emb = (const float*)d_in[2];
    const float* item_emb = (const float*)d_in[3];
    float*       out      = (float*)d_out;

    const int B       = in_sizes[0];
    const int n_items = in_sizes[3] / K_DIM;   // item_emb is (K, N)

    const int waves  = (B + 15) / 16;          // 16 outputs per wave
    const int blocks = (waves + 7) / 8;        // 8 waves per 256-thread block

    matcf_wmma_kernel<<<blocks, 256, 0, stream>>>(
        user, item, user_emb, item_emb, out, B, n_items);
}